// FrechetDistance_3762391351368
// MI455X (gfx1250) — compile-verified
//
#include <hip/hip_runtime.h>
#include <math.h>

// Fréchet distance, MI455X (gfx1250, wave32).
// One workgroup (256 threads = 8 waves) per batch element.
// Phase A: cost strip (16 rows x 256 cols) via V_WMMA_F32_16X16X4_F32:
//   cost = |x|^2 + |y|^2 - 2 x.y  ->  A=[x0 x1 x2 1], B=[-2y0 -2y1 -2y2 |y|^2],
//   C = |x_row|^2 broadcast. One WMMA per 16x16 tile, 16 tiles per strip.
// Phase B: DP rows as parallel scans. Row step v_j = max(c_j, min(m_j, v_{j-1}))
//   == clamp(v_{j-1}, c_j, max(c_j, m_j)); clamp composition is associative:
//   (B o A).lo = clamp(A.lo, B.lo, B.hi), (B o A).hi = clamp(A.hi, B.lo, B.hi).

typedef __attribute__((ext_vector_type(2))) float v2f;
typedef __attribute__((ext_vector_type(8))) float v8f;

#define NPTS    256
#define TILE    16
#define NSTRIPS 16

__device__ __forceinline__ float clampf(float v, float lo, float hi) {
    return fminf(hi, fmaxf(lo, v));
}

__global__ __launch_bounds__(256)
void frechet_wmma_kernel(const float* __restrict__ x,
                         const float* __restrict__ y,
                         float* __restrict__ out)
{
    __shared__ float nxS[NPTS];           // |x_i|^2 for all 256 rows
    __shared__ float costS[TILE * NPTS];  // current 16x256 cost strip
    __shared__ float prevS[NPTS];         // previous DP row
    __shared__ float waggLo[8], waggHi[8];// per-wave scan aggregates

    const int   b    = blockIdx.x;
    const int   tid  = threadIdx.x;
    const int   wave = tid >> 5;
    const int   lane = tid & 31;
    const float INF  = __builtin_inff();

    const float* xb = x + (size_t)b * NPTS * 3;
    const float* yb = y + (size_t)b * NPTS * 3;

    // one-time: x row norms + DP boundary row (all INF; (-1,-1) diag handled below)
    {
        float x0 = xb[tid * 3 + 0], x1 = xb[tid * 3 + 1], x2 = xb[tid * 3 + 2];
        nxS[tid]  = x0 * x0 + x1 * x1 + x2 * x2;
        prevS[tid] = INF;
    }
    __syncthreads();

    const int M      = lane & 15;  // matrix row (A) / col (B,C) index
    const int hiHalf = lane >> 4;  // 0: lanes 0-15 (K0,K1), 1: lanes 16-31 (K2,K3)

    for (int s = 0; s < NSTRIPS; ++s) {
        // ---------------- Phase A: WMMA cost strip ----------------
        const int xrow = TILE * s + M;
        float ax0 = xb[xrow * 3 + 0], ax1 = xb[xrow * 3 + 1], ax2 = xb[xrow * 3 + 2];
        v2f a;
        a.x = hiHalf ? ax2  : ax0;   // K2 : K0
        a.y = hiHalf ? 1.0f : ax1;   // K3 : K1
        v8f cacc;
        #pragma unroll
        for (int r = 0; r < 8; ++r)  // C(M=r+8*hiHalf, N) = |x_M|^2
            cacc[r] = nxS[TILE * s + r + 8 * hiHalf];

        #pragma unroll
        for (int k = 0; k < 2; ++k) {         // each wave owns 2 column tiles
            const int t    = 2 * wave + k;    // tile 0..15 (uniform per wave)
            const int yrow = TILE * t + M;
            float y0 = yb[yrow * 3 + 0], y1 = yb[yrow * 3 + 1], y2 = yb[yrow * 3 + 2];
            float ny = y0 * y0 + y1 * y1 + y2 * y2;
            v2f bb;
            bb.x = hiHalf ? (-2.0f * y2) : (-2.0f * y0);  // K2 : K0
            bb.y = hiHalf ? ny           : (-2.0f * y1);  // K3 : K1
            v8f d = __builtin_amdgcn_wmma_f32_16x16x4_f32(
                false, a, false, bb, (short)0, cacc, false, false);
            #pragma unroll
            for (int r = 0; r < 8; ++r)
                costS[(r + 8 * hiHalf) * NPTS + TILE * t + M] = d[r];
        }
        __syncthreads();

        // ---------------- Phase B: 16 DP rows via clamp-scan ----------------
        for (int rr = 0; rr < TILE; ++rr) {
            const int j = tid;
            float c  = fmaxf(costS[rr * NPTS + j], 0.0f);  // cost >= 0 exactly
            float up = prevS[j];
            float dg = (j > 0) ? prevS[j - 1] : INF;
            float m  = fminf(up, dg);
            if (s == 0 && rr == 0 && j == 0) m = 0.0f;     // D[-1][-1] = 0
            float lo = c;
            float hi = fmaxf(c, m);

            // intra-wave inclusive scan of clamp composition (wave32: 5 steps)
            #pragma unroll
            for (int off = 1; off < 32; off <<= 1) {
                float plo = __shfl_up(lo, off);
                float phi = __shfl_up(hi, off);
                if (lane >= off) {
                    float nlo = clampf(plo, lo, hi);
                    float nhi = clampf(phi, lo, hi);
                    lo = nlo;
                    hi = nhi;
                }
            }
            if (lane == 31) { waggLo[wave] = lo; waggHi[wave] = hi; }
            __syncthreads();

            // cross-wave prefix: apply earlier waves' composed clamps to v=INF,
            // then my own composed clamp.  Only the hi-chain is needed.
            float phi = INF;
            for (int w2 = 0; w2 < wave; ++w2)
                phi = clampf(phi, waggLo[w2], waggHi[w2]);
            float val = clampf(phi, lo, hi);

            prevS[j] = val;     // all prevS reads happened before the wagg barrier
            __syncthreads();    // also protects wagg for the next row
        }
    }

    if (tid == 0) out[b] = prevS[NPTS - 1];
}

extern "C" void kernel_launch(void* const* d_in, const int* in_sizes, int n_in,
                              void* d_out, int out_size, void* d_ws, size_t ws_size,
                              hipStream_t stream) {
    (void)n_in; (void)d_ws; (void)ws_size; (void)out_size;
    const float* x = (const float*)d_in[0];
    const float* y = (const float*)d_in[1];
    float* out = (float*)d_out;
    const int B = in_sizes[0] / (NPTS * 3);  // 128
    frechet_wmma_kernel<<<dim3(B), dim3(256), 0, stream>>>(x, y, out);
}